// TwoBodyEquiGraphConv_39651138076970
// MI455X (gfx1250) — compile-verified
//
#include <hip/hip_runtime.h>

// ---------------- problem constants ----------------
#define NN 50000      // nodes
#define NE 500000     // edges
#define CUT 5.0f
#define PI_F 3.14159265358979f
#define WAVES 8
#define TPB 256       // 8 wave32 per block

// persistent grid sizes (weights staged once per block, reused across tiles)
#define EDGE_BLOCKS 512
#define NODE_BLOCKS 96

typedef float v2f __attribute__((ext_vector_type(2)));
typedef float v8f __attribute__((ext_vector_type(8)));

// wave-local LDS ordering (CDNA5 split counters)
#define DS_WAIT() asm volatile("s_wait_dscnt 0" ::: "memory")

// ---------------- WMMA helpers (f32 16x16x4) ----------------
__device__ __forceinline__ v8f wmma4(v2f a, v2f b, v8f c) {
  return __builtin_amdgcn_wmma_f32_16x16x4_f32(false, a, false, b, (short)0, c, false, false);
}

__device__ __forceinline__ v8f splat8(float x) {
  v8f v;
#pragma unroll
  for (int r = 0; r < 8; ++r) v[r] = x;
  return v;
}

// B operand: rows k0..k0+3 of row-major W[K x NC], col block n0.
// lanes 0-15: K=k0,k0+1 / N=lane ; lanes 16-31: K=k0+2,k0+3 / N=lane-16
__device__ __forceinline__ v2f ldB(const float* W, int NC, int k0, int n0, int lane) {
  int kb  = k0 + ((lane >> 4) << 1);
  int col = n0 + (lane & 15);
  v2f b;
  b.x = W[kb * NC + col];
  b.y = W[(kb + 1) * NC + col];
  return b;
}

// A operand chunks for a 16x64 row-major tile in LDS
__device__ __forceinline__ void loadA64(v2f* a, const float* T, int lane) {
  int row = lane & 15, hi = lane >> 4;
#pragma unroll
  for (int kc = 0; kc < 16; ++kc) {
    int kb = kc * 4 + hi * 2;
    a[kc].x = T[row * 64 + kb];
    a[kc].y = T[row * 64 + kb + 1];
  }
}

// store a 16x64 C/D tile (4 N-blocks of v8f) to row-major LDS
__device__ __forceinline__ void storeC(float* T, const v8f* c, int lane) {
  int col = lane & 15, hi = lane >> 4;
#pragma unroll
  for (int nb = 0; nb < 4; ++nb)
#pragma unroll
    for (int r = 0; r < 8; ++r) T[(r + 8 * hi) * 64 + nb * 16 + col] = c[nb][r];
}

__device__ __forceinline__ v8f gemmK64(const v2f* a, const float* W, int NC, int n0,
                                       int lane, v8f acc) {
#pragma unroll
  for (int kc = 0; kc < 16; ++kc) acc = wmma4(a[kc], ldB(W, NC, kc * 4, n0, lane), acc);
  return acc;
}
__device__ __forceinline__ v8f gemmK128(const v2f* a, const float* W, int NC, int n0,
                                        int lane, v8f acc) {
#pragma unroll
  for (int kc = 0; kc < 32; ++kc) acc = wmma4(a[kc], ldB(W, NC, kc * 4, n0, lane), acc);
  return acc;
}

__device__ __forceinline__ float silu_f(float x) { return x / (1.0f + __expf(-x)); }

// shared-memory budgets (floats)
static constexpr int PW_STRIDE = 2176;  // per-wave: scr(1024)+tile(1024)+small(128)
static constexpr int EDGE_SM_FLOATS =
    128 * 64 + 3 * 64 * 64 + 64 * 192 + 448 + WAVES * PW_STRIDE;  // 50624 -> ~198KB
static constexpr int NODE_SM_FLOATS =
    64 * 192 + 128 * 64 + 2 * 64 * 128 + 384 + WAVES * PW_STRIDE; // 54656 -> ~214KB

// ---------------- workspace zero ----------------
__global__ void tbgc_zero_kernel(float* __restrict__ p, long n) {
  long i = (long)blockIdx.x * blockDim.x + threadIdx.x;
  long st = (long)gridDim.x * blockDim.x;
  for (; i < n; i += st) p[i] = 0.0f;
}

// ---------------- edge kernel: per-edge GEMM chain + scatter ----------------
__global__ void __launch_bounds__(TPB, 1) tbgc_edge_kernel(
    const float* __restrict__ node_s, const float* __restrict__ node_v,
    const float* __restrict__ edge_s, const float* __restrict__ edge_v,
    const float* __restrict__ dist, const float* __restrict__ vctr,
    const int* __restrict__ src, const int* __restrict__ dst,
    const float* __restrict__ W_nn, const float* __restrict__ b_nn,
    const float* __restrict__ W_ep, const float* __restrict__ b_ep,
    const float* __restrict__ W_g1, const float* __restrict__ b_g1,
    const float* __restrict__ W_g2, const float* __restrict__ b_g2,
    const float* __restrict__ W_ev, const float* __restrict__ b_ev,
    float* __restrict__ out_es, float* __restrict__ out_ev,
    float* __restrict__ deg, float* __restrict__ nes, float* __restrict__ nev) {
  extern __shared__ float sm[];
  float* sWnn = sm;                  // 128x64
  float* sWep = sWnn + 128 * 64;     // 64x64
  float* sWg1 = sWep + 64 * 64;      // 64x64
  float* sWg2 = sWg1 + 64 * 64;      // 64x64
  float* sWev = sWg2 + 64 * 64;      // 64x192
  float* sBia = sWev + 64 * 192;     // b_nn|b_ep|b_g1|b_g2 (4x64) + b_ev(192)
  float* sPW  = sBia + 448;

  const int tid = threadIdx.x;
  for (int i = tid; i < 128 * 64; i += TPB) sWnn[i] = W_nn[i];
  for (int i = tid; i < 64 * 64; i += TPB) { sWep[i] = W_ep[i]; sWg1[i] = W_g1[i]; sWg2[i] = W_g2[i]; }
  for (int i = tid; i < 64 * 192; i += TPB) sWev[i] = W_ev[i];
  for (int i = tid; i < 64; i += TPB) {
    sBia[i] = b_nn[i]; sBia[64 + i] = b_ep[i]; sBia[128 + i] = b_g1[i]; sBia[192 + i] = b_g2[i];
  }
  for (int i = tid; i < 192; i += TPB) sBia[256 + i] = b_ev[i];
  __syncthreads();

  const int wave = tid >> 5, lane = tid & 31;
  const int row = lane & 15, hi = lane >> 4, colb = lane & 15;

  float* scr = sPW + wave * PW_STRIDE;              // 16x64 staging (C<->A)
  float* esT = scr + 1024;                          // edge_s tile
  float* cT  = scr + 2048;                          // cutoff per edge
  float* vnT = scr + 2064;                          // 16x3 unit vectors
  int*   sidx = (int*)(scr + 2112);
  int*   didx = sidx + 16;

  const int nTiles = NE / 16;                       // NE % 16 == 0
  const int tstride = gridDim.x * WAVES;

  for (int t = blockIdx.x * WAVES + wave; t < nTiles; t += tstride) {
    const long base = (long)t * 16;

    if (lane < 16) {
      long e = base + lane;
      sidx[lane] = src[e];
      didx[lane] = dst[e];
      float dd = dist[e];
      cT[lane] = (dd < CUT) ? 0.5f * (__cosf(PI_F * dd / CUT) + 1.0f) : 0.0f;
      vnT[lane * 3 + 0] = vctr[e * 3 + 0];
      vnT[lane * 3 + 1] = vctr[e * 3 + 1];
      vnT[lane * 3 + 2] = vctr[e * 3 + 2];
    }
    for (int i = lane; i < 1024; i += 32) esT[i] = edge_s[base * 64 + i];  // coalesced
    DS_WAIT();

    if (lane < 16) atomicAdd(&deg[didx[lane]], 1.0f);

    // edge_message = ([ns_src | ns_dst] @ W_nn + b_nn) * (edge_s @ W_ep + b_ep)
    v8f em[4];
    {
      v2f aNN[32];
      const float* ps = node_s + (long)sidx[row] * 64;
      const float* pd = node_s + (long)didx[row] * 64;
#pragma unroll
      for (int kc = 0; kc < 16; ++kc) {
        int kb = kc * 4 + hi * 2;
        aNN[kc]      = *(const v2f*)(ps + kb);
        aNN[kc + 16] = *(const v2f*)(pd + kb);
      }
#pragma unroll
      for (int nb = 0; nb < 4; ++nb)
        em[nb] = gemmK128(aNN, sWnn, 64, nb * 16, lane, splat8(sBia[nb * 16 + colb]));
    }
    {
      v2f aE[16];
      loadA64(aE, esT, lane);
#pragma unroll
      for (int nb = 0; nb < 4; ++nb) {
        v8f ep = gemmK64(aE, sWep, 64, nb * 16, lane, splat8(sBia[64 + nb * 16 + colb]));
        em[nb] = em[nb] * ep;
      }
    }

    // h = silu(em @ W_g1 + b_g1)
    storeC(scr, em, lane);
    DS_WAIT();
    v8f esu[4];
    {
      v2f aT[16];
      loadA64(aT, scr, lane);
#pragma unroll
      for (int nb = 0; nb < 4; ++nb) {
        v8f h = gemmK64(aT, sWg1, 64, nb * 16, lane, splat8(sBia[128 + nb * 16 + colb]));
#pragma unroll
        for (int r = 0; r < 8; ++r) h[r] = silu_f(h[r]);
        esu[nb] = h;
      }
    }
    // es_upd = (h @ W_g2 + b_g2) * c
    storeC(scr, esu, lane);
    DS_WAIT();
    {
      v2f aT[16];
      loadA64(aT, scr, lane);
#pragma unroll
      for (int nb = 0; nb < 4; ++nb) {
        v8f o = gemmK64(aT, sWg2, 64, nb * 16, lane, splat8(sBia[192 + nb * 16 + colb]));
#pragma unroll
        for (int r = 0; r < 8; ++r) o[r] *= cT[r + 8 * hi];
        esu[nb] = o;
      }
    }

    // edge_s_out = es_upd + edge_s ; scatter-add n_es
#pragma unroll
    for (int nb = 0; nb < 4; ++nb) {
      int col = nb * 16 + colb;
#pragma unroll
      for (int r = 0; r < 8; ++r) {
        int M = r + 8 * hi;
        float v = esu[nb][r];
        long e = base + M;
        out_es[e * 64 + col] = v + esT[M * 64 + col];
        atomicAdd(&nes[(long)didx[M] * 64 + col], v);
      }
    }

    // vch = es_upd @ W_ev + b_ev -> nch|ech|rch ; ev_upd + outputs + scatter n_ev
    storeC(scr, esu, lane);
    DS_WAIT();
    {
      v2f aT[16];
      loadA64(aT, scr, lane);
#pragma unroll
      for (int fb = 0; fb < 4; ++fb) {
        int col = fb * 16 + colb;
        v8f nch = gemmK64(aT, sWev, 192, fb * 16, lane, splat8(sBia[256 + col]));
        v8f ech = gemmK64(aT, sWev, 192, 64 + fb * 16, lane, splat8(sBia[256 + 64 + col]));
        v8f rch = gemmK64(aT, sWev, 192, 128 + fb * 16, lane, splat8(sBia[256 + 128 + col]));
#pragma unroll
        for (int d = 0; d < 3; ++d) {
#pragma unroll
          for (int r = 0; r < 8; ++r) {
            int M = r + 8 * hi;
            long e = base + M;
            float evv = edge_v[e * 192 + d * 64 + col];
            float nvs = node_v[(long)sidx[M] * 192 + d * 64 + col];
            float ev = (nvs * nch[r] + evv * ech[r] + vnT[M * 3 + d] * rch[r]) * cT[M];
            out_ev[e * 192 + d * 64 + col] = ev + evv;
            atomicAdd(&nev[(long)didx[M] * 192 + d * 64 + col], ev);
          }
        }
      }
    }
  }
}

// ---------------- node kernel: mean, node GEMMs, LN + CoorsNorm ----------------
__global__ void __launch_bounds__(TPB, 1) tbgc_node_kernel(
    const float* __restrict__ node_s, const float* __restrict__ node_v,
    const float* __restrict__ W_nvout, const float* __restrict__ W_nvc,
    const float* __restrict__ b_nvc, const float* __restrict__ W_nvp,
    const float* __restrict__ W_nsp, const float* __restrict__ b_nsp,
    const float* __restrict__ ln_g, const float* __restrict__ ln_b,
    const float* __restrict__ cn_s,
    const float* __restrict__ deg, const float* __restrict__ nes,
    const float* __restrict__ nev,
    float* __restrict__ out_ns, float* __restrict__ out_nv) {
  extern __shared__ float sm[];
  float* sWvo = sm;                 // 64x192
  float* sWvc = sWvo + 64 * 192;    // 128x64
  float* sWvp = sWvc + 128 * 64;    // 64x128
  float* sWsp = sWvp + 64 * 128;    // 64x128
  float* sBia = sWsp + 64 * 128;    // b_nvc(64) b_nsp(128) gamma(64) beta(64) cn(64)
  float* sPW  = sBia + 384;

  const int tid = threadIdx.x;
  for (int i = tid; i < 64 * 192; i += TPB) sWvo[i] = W_nvout[i];
  for (int i = tid; i < 128 * 64; i += TPB) sWvc[i] = W_nvc[i];
  for (int i = tid; i < 64 * 128; i += TPB) { sWvp[i] = W_nvp[i]; sWsp[i] = W_nsp[i]; }
  for (int i = tid; i < 64; i += TPB) {
    sBia[i] = b_nvc[i]; sBia[192 + i] = ln_g[i]; sBia[256 + i] = ln_b[i]; sBia[320 + i] = cn_s[i];
  }
  for (int i = tid; i < 128; i += TPB) sBia[64 + i] = b_nsp[i];
  __syncthreads();

  const int wave = tid >> 5, lane = tid & 31;
  const int row = lane & 15, hi = lane >> 4, colb = lane & 15;

  float* scr  = sPW + wave * PW_STRIDE;
  float* nesT = scr + 1024;
  float* ideg = scr + 2048;  // 16
  float* muT  = scr + 2064;  // 16
  float* rsT  = scr + 2080;  // 16

  const int nTiles = NN / 16;                     // NN % 16 == 0
  const int tstride = gridDim.x * WAVES;

  for (int t = blockIdx.x * WAVES + wave; t < nTiles; t += tstride) {
    const long base = (long)t * 16;

    if (lane < 16) ideg[lane] = 1.0f / fmaxf(deg[base + lane], 1.0f);
    DS_WAIT();
    for (int i = lane; i < 1024; i += 32) nesT[i] = nes[base * 64 + i] * ideg[i >> 6];
    DS_WAIT();

    // pass 1: ||nv_o3||_d   (nv_o3 = n_ev @ W_nvout[:,128:192])
    v8f nrm2[4] = {splat8(0.f), splat8(0.f), splat8(0.f), splat8(0.f)};
    v2f aEV[16];
#pragma unroll
    for (int d = 0; d < 3; ++d) {
      const float* pv = nev + (base + row) * 192 + d * 64;
      float idg = ideg[row];
#pragma unroll
      for (int kc = 0; kc < 16; ++kc) {
        int kb = kc * 4 + hi * 2;
        v2f a = *(const v2f*)(pv + kb);
        aEV[kc].x = a.x * idg; aEV[kc].y = a.y * idg;
      }
#pragma unroll
      for (int nb = 0; nb < 4; ++nb) {
        v8f o3 = gemmK64(aEV, sWvo, 192, 128 + nb * 16, lane, splat8(0.f));
        nrm2[nb] += o3 * o3;
      }
    }
#pragma unroll
    for (int nb = 0; nb < 4; ++nb)
#pragma unroll
      for (int r = 0; r < 8; ++r) scr[(r + 8 * hi) * 64 + nb * 16 + colb] = sqrtf(nrm2[nb][r]);
    DS_WAIT();

    // v_channel = [n_es | norm] @ W_nvc + b_nvc
    v8f vc[4];
    {
      v2f aV[32];
      loadA64(aV, nesT, lane);
      loadA64(aV + 16, scr, lane);
#pragma unroll
      for (int nb = 0; nb < 4; ++nb)
        vc[nb] = gemmK128(aV, sWvc, 64, nb * 16, lane, splat8(sBia[nb * 16 + colb]));
    }

    // pass 2: nv_upd = o1 * vc + o2
    v8f nvu[3][4];
#pragma unroll
    for (int d = 0; d < 3; ++d) {
      const float* pv = nev + (base + row) * 192 + d * 64;
      float idg = ideg[row];
#pragma unroll
      for (int kc = 0; kc < 16; ++kc) {
        int kb = kc * 4 + hi * 2;
        v2f a = *(const v2f*)(pv + kb);
        aEV[kc].x = a.x * idg; aEV[kc].y = a.y * idg;
      }
#pragma unroll
      for (int nb = 0; nb < 4; ++nb) {
        v8f o1 = gemmK64(aEV, sWvo, 192, nb * 16, lane, splat8(0.f));
        v8f o2 = gemmK64(aEV, sWvo, 192, 64 + nb * 16, lane, splat8(0.f));
        nvu[d][nb] = o1 * vc[nb] + o2;
      }
    }

    // pass 3: nv_dot = sum_d (nv_upd @ W_nvp)[:, :64] * [:, 64:]
    v8f dotv[4] = {splat8(0.f), splat8(0.f), splat8(0.f), splat8(0.f)};
#pragma unroll
    for (int d = 0; d < 3; ++d) {
      storeC(scr, nvu[d], lane);
      DS_WAIT();
      v2f aT[16];
      loadA64(aT, scr, lane);
#pragma unroll
      for (int fb = 0; fb < 4; ++fb) {
        v8f p1 = gemmK64(aT, sWvp, 128, fb * 16, lane, splat8(0.f));
        v8f p2 = gemmK64(aT, sWvp, 128, 64 + fb * 16, lane, splat8(0.f));
        dotv[fb] += p1 * p2;
      }
    }

    // ns_upd = nv_dot * silu(nsp1) + silu(nsp2)
    v8f nsU[4];
    {
      v2f aS[16];
      loadA64(aS, nesT, lane);
#pragma unroll
      for (int fb = 0; fb < 4; ++fb) {
        int col = fb * 16 + colb;
        v8f s1 = gemmK64(aS, sWsp, 128, fb * 16, lane, splat8(sBia[64 + col]));
        v8f s2 = gemmK64(aS, sWsp, 128, 64 + fb * 16, lane, splat8(sBia[64 + 64 + col]));
#pragma unroll
        for (int r = 0; r < 8; ++r) { s1[r] = silu_f(s1[r]); s2[r] = silu_f(s2[r]); }
        nsU[fb] = dotv[fb] * s1 + s2;
      }
    }

    // residual + LayerNorm over F
#pragma unroll
    for (int fb = 0; fb < 4; ++fb) {
      int col = fb * 16 + colb;
#pragma unroll
      for (int r = 0; r < 8; ++r) {
        int M = r + 8 * hi;
        nsU[fb][r] += node_s[(base + M) * 64 + col];
        scr[M * 64 + col] = nsU[fb][r];
      }
    }
    DS_WAIT();
    if (lane < 16) {
      float s = 0.f, q = 0.f;
#pragma unroll
      for (int j = 0; j < 64; ++j) { float v = scr[lane * 64 + j]; s += v; q += v * v; }
      float mu = s * (1.0f / 64.0f);
      float var = q * (1.0f / 64.0f) - mu * mu;
      muT[lane] = mu;
      rsT[lane] = 1.0f / sqrtf(var + 1e-5f);
    }
    DS_WAIT();
#pragma unroll
    for (int fb = 0; fb < 4; ++fb) {
      int col = fb * 16 + colb;
#pragma unroll
      for (int r = 0; r < 8; ++r) {
        int M = r + 8 * hi;
        out_ns[(base + M) * 64 + col] =
            (nsU[fb][r] - muT[M]) * rsT[M] * sBia[192 + col] + sBia[256 + col];
      }
    }

    // node_v residual + CoorsNorm
    v8f vn2[4] = {splat8(0.f), splat8(0.f), splat8(0.f), splat8(0.f)};
#pragma unroll
    for (int d = 0; d < 3; ++d)
#pragma unroll
      for (int fb = 0; fb < 4; ++fb) {
        int col = fb * 16 + colb;
#pragma unroll
        for (int r = 0; r < 8; ++r) {
          int M = r + 8 * hi;
          float x = nvu[d][fb][r] + node_v[(base + M) * 192 + d * 64 + col];
          nvu[d][fb][r] = x;
          vn2[fb][r] += x * x;
        }
      }
#pragma unroll
    for (int fb = 0; fb < 4; ++fb) {
      int col = fb * 16 + colb;
#pragma unroll
      for (int r = 0; r < 8; ++r) {
        int M = r + 8 * hi;
        float sc = sBia[320 + col] / fmaxf(sqrtf(vn2[fb][r]), 1e-8f);
#pragma unroll
        for (int d = 0; d < 3; ++d)
          out_nv[(base + M) * 192 + d * 64 + col] = nvu[d][fb][r] * sc;
      }
    }
  }
}

// ---------------- host launch ----------------
extern "C" void kernel_launch(void* const* d_in, const int* in_sizes, int n_in,
                              void* d_out, int out_size, void* d_ws, size_t ws_size,
                              hipStream_t stream) {
  (void)in_sizes; (void)n_in; (void)out_size; (void)ws_size;
  const float* node_s = (const float*)d_in[0];
  const float* node_v = (const float*)d_in[1];
  const float* edge_s = (const float*)d_in[2];
  const float* edge_v = (const float*)d_in[3];
  const float* dist   = (const float*)d_in[4];
  const float* vctr   = (const float*)d_in[5];
  const int*   src    = (const int*)d_in[6];
  const int*   dst    = (const int*)d_in[7];
  const float* W_nn = (const float*)d_in[8];  const float* b_nn = (const float*)d_in[9];
  const float* W_ep = (const float*)d_in[10]; const float* b_ep = (const float*)d_in[11];
  const float* W_g1 = (const float*)d_in[12]; const float* b_g1 = (const float*)d_in[13];
  const float* W_g2 = (const float*)d_in[14]; const float* b_g2 = (const float*)d_in[15];
  const float* W_ev = (const float*)d_in[16]; const float* b_ev = (const float*)d_in[17];
  const float* W_nvout = (const float*)d_in[18];
  const float* W_nvc = (const float*)d_in[19]; const float* b_nvc = (const float*)d_in[20];
  const float* W_nvp = (const float*)d_in[21];
  const float* W_nsp = (const float*)d_in[22]; const float* b_nsp = (const float*)d_in[23];
  const float* ln_g = (const float*)d_in[24]; const float* ln_b = (const float*)d_in[25];
  const float* cn_s = (const float*)d_in[26];

  float* out = (float*)d_out;
  float* out_ns = out;
  float* out_nv = out_ns + (long)NN * 64;
  float* out_es = out_nv + (long)NN * 192;
  float* out_ev = out_es + (long)NE * 64;

  float* ws  = (float*)d_ws;                 // deg[NN] | nes[NN*64] | nev[NN*192]
  float* deg = ws;
  float* nes = deg + NN;
  float* nev = nes + (long)NN * 64;

  tbgc_zero_kernel<<<768, 256, 0, stream>>>(ws, (long)NN * 257);

  size_t shE = (size_t)EDGE_SM_FLOATS * sizeof(float);
  size_t shN = (size_t)NODE_SM_FLOATS * sizeof(float);
  hipFuncSetAttribute(reinterpret_cast<const void*>(tbgc_edge_kernel),
                      hipFuncAttributeMaxDynamicSharedMemorySize, (int)shE);
  hipFuncSetAttribute(reinterpret_cast<const void*>(tbgc_node_kernel),
                      hipFuncAttributeMaxDynamicSharedMemorySize, (int)shN);

  tbgc_edge_kernel<<<EDGE_BLOCKS, TPB, shE, stream>>>(
      node_s, node_v, edge_s, edge_v, dist, vctr, src, dst,
      W_nn, b_nn, W_ep, b_ep, W_g1, b_g1, W_g2, b_g2, W_ev, b_ev,
      out_es, out_ev, deg, nes, nev);

  tbgc_node_kernel<<<NODE_BLOCKS, TPB, shN, stream>>>(
      node_s, node_v, W_nvout, W_nvc, b_nvc, W_nvp, W_nsp, b_nsp,
      ln_g, ln_b, cn_s, deg, nes, nev, out_ns, out_nv);
}